// GAT_44805098832217
// MI455X (gfx1250) — compile-verified
//
#include <hip/hip_runtime.h>
#include <hip/hip_bf16.h>

#define N_NODES 100000
#define E_EDGES 1600000
#define IN_FEAT 256
#define HID     16
#define HEADS   8
#define OUT_FEAT 64
#define NEG_SLOPE 0.2f
#define NEG_INF_KEY ((int)0x807FFFFF)   // fkey(-inf)

typedef float v2f __attribute__((ext_vector_type(2)));
typedef float v8f __attribute__((ext_vector_type(8)));
typedef unsigned int v4u __attribute__((ext_vector_type(4)));
typedef int v4i __attribute__((ext_vector_type(4)));
typedef int v8i __attribute__((ext_vector_type(8)));

// ---- order-preserving float<->int key for atomicMax on signed int ----
__device__ __forceinline__ int fkey(float f) {
    int fi = __float_as_int(f);
    return fi >= 0 ? fi : (fi ^ 0x7FFFFFFF);
}
__device__ __forceinline__ float funkey(int k) {
    return __int_as_float(k >= 0 ? k : (k ^ 0x7FFFFFFF));
}

__device__ __forceinline__ void edge_ends(const long long* __restrict__ ei,
                                          int e, int& s, int& d) {
    if (e < E_EDGES) { s = (int)ei[e]; d = (int)ei[(long long)E_EDGES + e]; }
    else             { s = e - E_EDGES; d = s; }   // appended self-loops
}

// ---- Tensor Data Mover: DMA a [rows x cols] fp32 tile (row stride in elems)
// from global memory into LDS. D# built per CDNA5 ISA ch.8 (2D tensor:
// group0 = count/lds_addr/global_addr/type, group1 = data_size/dims/strides;
// groups 2,3 (and the clang-23 extra group) zero for a 2D tile).
__device__ __forceinline__ void tdm_load_tile_f32(const float* gsrc, void* lds_dst,
                                                  unsigned rows, unsigned cols,
                                                  unsigned row_stride_elems) {
    unsigned long long ga = (unsigned long long)(uintptr_t)gsrc;
    unsigned lds_off = (unsigned)(uintptr_t)lds_dst;
    v4u g0;
    g0.x = 1u;                                                  // count=1, user mode
    g0.y = lds_off;                                             // lds_addr [63:32]
    g0.z = (unsigned)ga;                                        // global_addr [95:64]
    g0.w = (unsigned)((ga >> 32) & 0x01FFFFFFu) | (2u << 30);   // addr[56:32] | type=2
    v8i g1;
    g1[0] = (int)(2u << 16);                                    // data_size=2 (4 bytes)
    g1[1] = (int)((cols & 0xFFFFu) << 16);                      // tensor_dim0 [79:48] lo
    g1[2] = (int)(((cols >> 16) & 0xFFFFu) |                    // tensor_dim0 hi
                  ((rows & 0xFFFFu) << 16));                    // tensor_dim1 [111:80] lo
    g1[3] = (int)(((rows >> 16) & 0xFFFFu) |                    // tensor_dim1 hi
                  ((cols & 0xFFFFu) << 16));                    // tile_dim0 [127:112]
    g1[4] = (int)(rows & 0xFFFFu);                              // tile_dim1; tile_dim2=0
    g1[5] = (int)row_stride_elems;                              // dim0_stride [207:160] lo
    g1[6] = 0;                                                  // dim0_stride hi | dim1_stride lo
    g1[7] = 0;
    v4i gz = {0, 0, 0, 0};
#if defined(__clang_major__) && (__clang_major__ >= 23)
    v8i gz8 = {0, 0, 0, 0, 0, 0, 0, 0};
    __builtin_amdgcn_tensor_load_to_lds(g0, g1, gz, gz, gz8, 0);
#else
    __builtin_amdgcn_tensor_load_to_lds(g0, g1, gz, gz, 0);
#endif
}

// ---------------- fills ----------------
__global__ void fill_f32(float* __restrict__ p, float v, long long n) {
    long long i = (long long)blockIdx.x * blockDim.x + threadIdx.x;
    if (i < n) p[i] = v;
}
__global__ void fill_i32(int* __restrict__ p, int v, long long n) {
    long long i = (long long)blockIdx.x * blockDim.x + threadIdx.x;
    if (i < n) p[i] = v;
}

// ---------------- WMMA GEMM: C[nrows,NCOL] = A[nrows,K] * B[K,NCOL] --------
// One block = 16 output rows; wave w owns column tile [16w, 16w+16).
// A tile staged in LDS by the Tensor Data Mover (TDM); compute uses
// V_WMMA_F32_16X16X4_F32 (fp32 in, fp32 accumulate, K stepped by 4).
template<int K, int NCOL>
__global__ __launch_bounds__((NCOL / 16) * 32)
void gemm_wmma_f32(const float* __restrict__ A, const float* __restrict__ B,
                   float* __restrict__ C, int nrows) {
    __shared__ float As[16 * K];
    const int row0 = blockIdx.x * 16;
    if (row0 >= nrows) return;

    // Wave 0 issues one TDM descriptor for the whole 16xK tile, then waits
    // on TENSORcnt; the workgroup barrier publishes the LDS tile to all waves.
    if ((threadIdx.x >> 5) == 0) {
        tdm_load_tile_f32(A + (long long)row0 * K, As, 16u, (unsigned)K, (unsigned)K);
        __builtin_amdgcn_s_wait_tensorcnt(0);
    }
    __syncthreads();

    const int lane  = threadIdx.x & 31;
    const int wave  = threadIdx.x >> 5;
    const int col0  = wave * 16;
    const int ln15  = lane & 15;    // A: M index; B: N index
    const int khalf = lane >> 4;    // 0 -> K pair {0,1}; 1 -> K pair {2,3}

    __builtin_prefetch(B + col0, 0, 3);   // pull this wave's B slab toward L2

    v8f acc = {};
    for (int k0 = 0; k0 < K; k0 += 4) {
        const int kk = k0 + 2 * khalf;
        v2f a, b;
        a.x = As[ln15 * K + kk];
        a.y = As[ln15 * K + kk + 1];
        b.x = B[(long long)kk       * NCOL + col0 + ln15];
        b.y = B[(long long)(kk + 1) * NCOL + col0 + ln15];
        acc = __builtin_amdgcn_wmma_f32_16x16x4_f32(
            /*neg_a=*/false, a, /*neg_b=*/false, b,
            /*c_mod=*/(short)0, acc, /*reuse_a=*/false, /*reuse_b=*/false);
    }
    #pragma unroll
    for (int r = 0; r < 8; ++r) {
        const int m = r + 8 * khalf;             // C/D layout: VGPR r -> M=r / M=r+8
        C[(long long)(row0 + m) * NCOL + col0 + ln15] = acc[r];
    }
}

// ---------------- attention dot products: a = sum_f xl[n,h,f]*att[h,f] ----
__global__ void att_dot_kernel(const float* __restrict__ xl,
                               const float* __restrict__ att_s,
                               const float* __restrict__ att_d,
                               float* __restrict__ as_, float* __restrict__ ad_,
                               int n, int heads, int fdim) {
    long long i = (long long)blockIdx.x * blockDim.x + threadIdx.x; // node*heads + h
    if (i >= (long long)n * heads) return;
    const int h = (int)(i % heads);
    const float* v  = xl + i * fdim;
    const float* ps = att_s + h * fdim;
    const float* pd = att_d + h * fdim;
    float ss = 0.f, sd = 0.f;
    for (int f = 0; f < fdim; ++f) { ss += v[f] * ps[f]; sd += v[f] * pd[f]; }
    as_[i] = ss; ad_[i] = sd;
}

// ---------------- edge pass 1: segment max over dst ----------------
template<int HT>
__global__ void edge_max_kernel(const long long* __restrict__ ei,
                                const float* __restrict__ as_,
                                const float* __restrict__ ad_,
                                int* __restrict__ mkey, int total_edges) {
    long long tid = (long long)blockIdx.x * blockDim.x + threadIdx.x;
    if (tid >= (long long)total_edges * HT) return;
    const int e = (int)(tid / HT), h = (int)(tid % HT);
    int s, d; edge_ends(ei, e, s, d);
    float v = as_[s * HT + h] + ad_[d * HT + h];
    v = v > 0.f ? v : NEG_SLOPE * v;
    atomicMax(mkey + d * HT + h, fkey(v));
}

// ---------------- edge pass 2: segment sum of exp(e - m) ----------------
template<int HT>
__global__ void edge_sum_kernel(const long long* __restrict__ ei,
                                const float* __restrict__ as_,
                                const float* __restrict__ ad_,
                                const int* __restrict__ mkey,
                                float* __restrict__ den, int total_edges) {
    long long tid = (long long)blockIdx.x * blockDim.x + threadIdx.x;
    if (tid >= (long long)total_edges * HT) return;
    const int e = (int)(tid / HT), h = (int)(tid % HT);
    int s, d; edge_ends(ei, e, s, d);
    const int idx = d * HT + h;
    float v = as_[s * HT + h] + ad_[idx];
    v = v > 0.f ? v : NEG_SLOPE * v;
    atomicAdd(den + idx, __expf(v - funkey(mkey[idx])));
}

// ---------------- edge pass 3: out[dst] += alpha * xl[src] --------------
// One thread per (edge, 4-channel group): alpha computed once per group
// (HID and OUT_FEAT are multiples of 4, so a group never crosses heads);
// the gathered features move as a single b128 load.
template<int HT, int FD>
__global__ void edge_scatter_kernel(const long long* __restrict__ ei,
                                    const float* __restrict__ as_,
                                    const float* __restrict__ ad_,
                                    const int* __restrict__ mkey,
                                    const float* __restrict__ den,
                                    const float* __restrict__ xl,
                                    float* __restrict__ out, int total_edges) {
    const int F = HT * FD;
    const int Q = F / 4;
    long long tid = (long long)blockIdx.x * blockDim.x + threadIdx.x;
    if (tid >= (long long)total_edges * Q) return;
    const int e  = (int)(tid / Q);
    const int q  = (int)(tid % Q);
    const int c0 = q * 4;
    const int h  = c0 / FD;
    int s, d; edge_ends(ei, e, s, d);
    const int idx = d * HT + h;
    float v = as_[s * HT + h] + ad_[idx];
    v = v > 0.f ? v : NEG_SLOPE * v;
    const float alpha = __expf(v - funkey(mkey[idx])) / den[idx];
    const float4 xv = *(const float4*)(xl + (long long)s * F + c0);
    float* op = out + (long long)d * F + c0;
    atomicAdd(op + 0, alpha * xv.x);
    atomicAdd(op + 1, alpha * xv.y);
    atomicAdd(op + 2, alpha * xv.z);
    atomicAdd(op + 3, alpha * xv.w);
}

// ---------------- H = elu(acc + b1) (in place) ----------------
__global__ void elu_bias_kernel(float* __restrict__ h, const float* __restrict__ b,
                                int fwidth, long long n) {
    long long i = (long long)blockIdx.x * blockDim.x + threadIdx.x;
    if (i >= n) return;
    float v = h[i] + b[(int)(i % fwidth)];
    h[i] = v > 0.f ? v : expm1f(v);
}

// ---------------- out += b2 ----------------
__global__ void bias_add_kernel(float* __restrict__ out, const float* __restrict__ b,
                                int fwidth, long long n) {
    long long i = (long long)blockIdx.x * blockDim.x + threadIdx.x;
    if (i >= n) return;
    out[i] += b[(int)(i % fwidth)];
}

static inline unsigned cdiv_ll(long long a, long long b) {
    return (unsigned)((a + b - 1) / b);
}

extern "C" void kernel_launch(void* const* d_in, const int* in_sizes, int n_in,
                              void* d_out, int out_size, void* d_ws, size_t ws_size,
                              hipStream_t stream) {
    (void)in_sizes; (void)n_in; (void)out_size; (void)ws_size;
    const float*     x   = (const float*)d_in[0];
    const long long* ei  = (const long long*)d_in[1];
    const float*     W1  = (const float*)d_in[2];
    const float*     as1 = (const float*)d_in[3];
    const float*     ad1 = (const float*)d_in[4];
    const float*     b1  = (const float*)d_in[5];
    const float*     W2  = (const float*)d_in[6];
    const float*     as2 = (const float*)d_in[7];
    const float*     ad2 = (const float*)d_in[8];
    const float*     b2  = (const float*)d_in[9];
    float*           out = (float*)d_out;

    const long long Nl = N_NODES;
    const int F1 = HEADS * HID;          // 128
    float* XL1   = (float*)d_ws;         // N*128
    float* H     = XL1 + Nl * F1;        // N*128 (accumulator, then elu in place)
    float* ASRC1 = H + Nl * F1;          // N*8
    float* ADST1 = ASRC1 + Nl * HEADS;   // N*8
    int*   MK    = (int*)(ADST1 + Nl * HEADS);  // N*8
    float* DEN   = (float*)(MK + Nl * HEADS);   // N*8
    // layer-2 reuses the XL1 region (free after layer-1 aggregation)
    float* XL2   = XL1;                  // N*64
    float* ASRC2 = XL1 + Nl * OUT_FEAT;  // N
    float* ADST2 = ASRC2 + Nl;           // N
    int*   MK2   = MK;                   // N
    float* DEN2  = DEN;                  // N

    const int TE = E_EDGES + N_NODES;    // edges + self-loops

    // ======== layer 1 ========
    fill_f32<<<cdiv_ll(Nl * F1, 256), 256, 0, stream>>>(H, 0.f, Nl * F1);
    fill_i32<<<cdiv_ll(Nl * HEADS, 256), 256, 0, stream>>>(MK, NEG_INF_KEY, Nl * HEADS);
    fill_f32<<<cdiv_ll(Nl * HEADS, 256), 256, 0, stream>>>(DEN, 0.f, Nl * HEADS);

    gemm_wmma_f32<IN_FEAT, 128><<<N_NODES / 16, 256, 0, stream>>>(x, W1, XL1, N_NODES);
    att_dot_kernel<<<cdiv_ll(Nl * HEADS, 256), 256, 0, stream>>>(
        XL1, as1, ad1, ASRC1, ADST1, N_NODES, HEADS, HID);

    edge_max_kernel<HEADS><<<cdiv_ll((long long)TE * HEADS, 256), 256, 0, stream>>>(
        ei, ASRC1, ADST1, MK, TE);
    edge_sum_kernel<HEADS><<<cdiv_ll((long long)TE * HEADS, 256), 256, 0, stream>>>(
        ei, ASRC1, ADST1, MK, DEN, TE);
    edge_scatter_kernel<HEADS, HID><<<cdiv_ll((long long)TE * (F1 / 4), 256), 256, 0, stream>>>(
        ei, ASRC1, ADST1, MK, DEN, XL1, H, TE);

    elu_bias_kernel<<<cdiv_ll(Nl * F1, 256), 256, 0, stream>>>(H, b1, F1, Nl * F1);

    // ======== layer 2 ========
    gemm_wmma_f32<128, OUT_FEAT><<<N_NODES / 16, 128, 0, stream>>>(H, W2, XL2, N_NODES);
    att_dot_kernel<<<cdiv_ll(Nl, 256), 256, 0, stream>>>(
        XL2, as2, ad2, ASRC2, ADST2, N_NODES, 1, OUT_FEAT);

    fill_i32<<<cdiv_ll(Nl, 256), 256, 0, stream>>>(MK2, NEG_INF_KEY, Nl);
    fill_f32<<<cdiv_ll(Nl, 256), 256, 0, stream>>>(DEN2, 0.f, Nl);
    fill_f32<<<cdiv_ll(Nl * OUT_FEAT, 256), 256, 0, stream>>>(out, 0.f, Nl * OUT_FEAT);

    edge_max_kernel<1><<<cdiv_ll((long long)TE, 256), 256, 0, stream>>>(
        ei, ASRC2, ADST2, MK2, TE);
    edge_sum_kernel<1><<<cdiv_ll((long long)TE, 256), 256, 0, stream>>>(
        ei, ASRC2, ADST2, MK2, DEN2, TE);
    edge_scatter_kernel<1, OUT_FEAT><<<cdiv_ll((long long)TE * (OUT_FEAT / 4), 256), 256, 0, stream>>>(
        ei, ASRC2, ADST2, MK2, DEN2, XL2, out, TE);

    bias_add_kernel<<<cdiv_ll(Nl * OUT_FEAT, 256), 256, 0, stream>>>(
        out, b2, OUT_FEAT, Nl * OUT_FEAT);
}